// DisenGCN_32160715112488
// MI455X (gfx1250) — compile-verified
//
#include <hip/hip_runtime.h>
#include <hip/hip_bf16.h>
#include <math.h>

// GCN forward for MI455X (gfx1250).
// - fp32 WMMA (V_WMMA_F32_16X16X4_F32) for node-feature GEMMs (exact fp32).
// - norm(s,d)=dinv[s]*dinv[d] factorized: src-side dinv fused into the WMMA
//   store, dst-side dinv fused into the finish pass -> edge scatter is a pure
//   vectorized gather + global_atomic_add_f32 (L2-resident: h fits in 192MB L2).

typedef float v2f __attribute__((ext_vector_type(2)));
typedef float v8f __attribute__((ext_vector_type(8)));

#define BN_EPS 1e-5f

// ---------------- utility ----------------
__global__ void k_zero(float* __restrict__ p, size_t n) {
  size_t i = (size_t)blockIdx.x * blockDim.x + threadIdx.x;
  size_t stride = (size_t)gridDim.x * blockDim.x;
  for (; i < n; i += stride) p[i] = 0.0f;
}

// ---------------- degree / norm ----------------
__global__ void k_deg_init(float* __restrict__ deg, int n) {
  int i = blockIdx.x * blockDim.x + threadIdx.x;
  if (i < n) deg[i] = 1.0f;  // self-loop contribution
}

__global__ void k_deg_count(const int* __restrict__ dst, float* __restrict__ deg, int E) {
  int e = blockIdx.x * blockDim.x + threadIdx.x;
  if (e < E) unsafeAtomicAdd(&deg[dst[e]], 1.0f);
}

__global__ void k_rsqrt_inplace(float* __restrict__ d, int n) {
  int i = blockIdx.x * blockDim.x + threadIdx.x;
  if (i < n) {
    float v = d[i];
    d[i] = (v > 0.0f) ? rsqrtf(v) : 0.0f;
  }
}

// ------- WMMA GEMM: C[MxF] = (A[MxK] @ B[KxF]) * dinv[m] (row-scaled) -------
// One wave computes a 16x16 output tile. ISA VGPR layouts (CDNA5, wave32):
//   A 16x4 : lane m=(lane&15), k = vgpr + 2*(lane>=16)
//   B 4x16 : lane n=(lane&15), k = vgpr + 2*(lane>=16)
//   C 16x16: lane n=(lane&15), m = vgpr + 8*(lane>=16)
__global__ void k_gemm_wmma_f32(const float* __restrict__ A, const float* __restrict__ B,
                                const float* __restrict__ dinv,
                                float* __restrict__ C, int M, int K, int F) {
  const int lane = threadIdx.x & 31;
  const int wave = (int)((blockIdx.x * blockDim.x + threadIdx.x) >> 5);
  const int nf = F >> 4;
  const int mTile = wave / nf;
  const int fTile = wave - mTile * nf;
  const int m0 = mTile << 4;
  if (m0 >= M) return;  // wave-uniform: EXEC stays all-ones for WMMA
  const int f0 = fTile << 4;
  const int row = lane & 15;
  const int kh2 = (lane >> 4) << 1;  // 0 or 2

  const float* arow = A + (size_t)(m0 + row) * K + kh2;
  const float* bcol = B + (size_t)kh2 * F + f0 + row;

  v8f acc = {0.f, 0.f, 0.f, 0.f, 0.f, 0.f, 0.f, 0.f};
  for (int k = 0; k < K; k += 4) {
    float2 av = *reinterpret_cast<const float2*>(arow + k);  // A[m][k+kh2 .. +1]
    v2f a; a.x = av.x; a.y = av.y;
    v2f b;
    b.x = bcol[(size_t)k * F];        // B[k + kh2    ][f]
    b.y = bcol[(size_t)(k + 1) * F];  // B[k + kh2 + 1][f]
    acc = __builtin_amdgcn_wmma_f32_16x16x4_f32(false, a, false, b,
                                                (short)0, acc, false, false);
  }
  // row-scale by dinv (src-side GCN norm factor) during writeback
  const float* dvp = dinv + m0 + (kh2 << 2);  // rows m0 + 8*khalf + r
  float* crow = C + (size_t)(m0 + (kh2 << 2)) * F + f0 + row;
#pragma unroll
  for (int r = 0; r < 8; ++r) crow[(size_t)r * F] = acc[r] * dvp[r];
}

// -------- edge scatter-add: one wave per edge, b128 gather + f32 atomics ----
// agg[d] += h_scaled[s]   (weights already folded into h_scaled / finish)
__global__ void k_scatter(const float* __restrict__ hsc, const int* __restrict__ src,
                          const int* __restrict__ dst,
                          float* __restrict__ agg, int E, int F) {
  int wave = (int)((blockIdx.x * blockDim.x + threadIdx.x) >> 5);
  int lane = threadIdx.x & 31;
  if (wave >= E) return;
  int s = src[wave], d = dst[wave];
  const float4* hrow = reinterpret_cast<const float4*>(hsc + (size_t)s * F);
  float* ad = agg + (size_t)d * F;
  for (int f4 = lane; 4 * f4 < F; f4 += 32) {
    float4 v = hrow[f4];
    unsafeAtomicAdd(ad + 4 * f4 + 0, v.x);
    unsafeAtomicAdd(ad + 4 * f4 + 1, v.y);
    unsafeAtomicAdd(ad + 4 * f4 + 2, v.z);
    unsafeAtomicAdd(ad + 4 * f4 + 3, v.w);
  }
}

// -- fused: dst-side norm + self-loop + bias + relu + BN-stats accumulation --
// v = relu((agg + h_scaled) * dinv[r] + b);  blockDim.x == F.
__global__ void k_finish_relu_stats(float* __restrict__ agg, const float* __restrict__ hsc,
                                    const float* __restrict__ dinv,
                                    const float* __restrict__ bias,
                                    float* __restrict__ stats,  // [0..F) sum, [F..2F) sumsq
                                    int Nn, int F, int rowsPerBlock) {
  int j = threadIdx.x;
  int r0 = blockIdx.x * rowsPerBlock;
  int r1 = min(r0 + rowsPerBlock, Nn);
  float bj = bias[j];
  float sum = 0.f, sumsq = 0.f;
  for (int r = r0; r < r1; ++r) {
    float di = dinv[r];
    size_t idx = (size_t)r * F + j;
    float v = (agg[idx] + hsc[idx]) * di + bj;
    v = fmaxf(v, 0.f);
    agg[idx] = v;
    sum += v;
    sumsq += v * v;
  }
  unsafeAtomicAdd(stats + j, sum);
  unsafeAtomicAdd(stats + F + j, sumsq);
}

// ---------------- BatchNorm apply (training-mode biased var) ----------------
__global__ void k_bn_apply(float* __restrict__ v, const float* __restrict__ stats,
                           const float* __restrict__ gamma, const float* __restrict__ beta,
                           int Nn, int F, float invN) {
  size_t total = (size_t)Nn * F;
  size_t i = (size_t)blockIdx.x * blockDim.x + threadIdx.x;
  size_t stride = (size_t)gridDim.x * blockDim.x;
  for (; i < total; i += stride) {
    int j = (int)(i & (size_t)(F - 1));  // F is a power of two (128 / 64)
    float mean = stats[j] * invN;
    float var = stats[F + j] * invN - mean * mean;
    v[i] = (v[i] - mean) * rsqrtf(var + BN_EPS) * gamma[j] + beta[j];
  }
}

// ---------------- layer 4: [Nx64] @ [64x2], row-scaled by dinv --------------
__global__ void k_gemm64x2(const float* __restrict__ h3, const float* __restrict__ W4,
                           const float* __restrict__ dinv,
                           float* __restrict__ h4, int Nn) {
  int i = blockIdx.x * blockDim.x + threadIdx.x;
  if (i >= Nn) return;
  const float* row = h3 + (size_t)i * 64;
  float o0 = 0.f, o1 = 0.f;
#pragma unroll 8
  for (int k = 0; k < 64; ++k) {
    float xv = row[k];
    o0 += xv * W4[2 * k];
    o1 += xv * W4[2 * k + 1];
  }
  float di = dinv[i];
  h4[2 * (size_t)i]     = o0 * di;
  h4[2 * (size_t)i + 1] = o1 * di;
}

__global__ void k_scatter2(const float* __restrict__ h4s, const int* __restrict__ src,
                           const int* __restrict__ dst,
                           float* __restrict__ agg, int E) {
  int e = blockIdx.x * blockDim.x + threadIdx.x;
  if (e >= E) return;
  int s = src[e], d = dst[e];
  float2 v = *reinterpret_cast<const float2*>(h4s + 2 * (size_t)s);
  unsafeAtomicAdd(agg + 2 * (size_t)d,     v.x);
  unsafeAtomicAdd(agg + 2 * (size_t)d + 1, v.y);
}

__global__ void k_final_lsm(const float* __restrict__ agg, const float* __restrict__ h4s,
                            const float* __restrict__ dinv, const float* __restrict__ b4,
                            float* __restrict__ out, int Nn) {
  int i = blockIdx.x * blockDim.x + threadIdx.x;
  if (i >= Nn) return;
  float di = dinv[i];
  float o0 = (agg[2 * (size_t)i]     + h4s[2 * (size_t)i])     * di + b4[0];
  float o1 = (agg[2 * (size_t)i + 1] + h4s[2 * (size_t)i + 1]) * di + b4[1];
  float m = fmaxf(o0, o1);
  float l = m + logf(expf(o0 - m) + expf(o1 - m));
  out[2 * (size_t)i]     = o0 - l;
  out[2 * (size_t)i + 1] = o1 - l;
}

// ---------------- orchestration ----------------
extern "C" void kernel_launch(void* const* d_in, const int* in_sizes, int n_in,
                              void* d_out, int out_size, void* d_ws, size_t ws_size,
                              hipStream_t stream) {
  const float* x   = (const float*)d_in[0];
  const int*   ei  = (const int*)d_in[1];
  const float* W1  = (const float*)d_in[2];
  const float* b1  = (const float*)d_in[3];
  const float* g1  = (const float*)d_in[4];
  const float* be1 = (const float*)d_in[5];
  const float* W2  = (const float*)d_in[6];
  const float* b2  = (const float*)d_in[7];
  const float* g2  = (const float*)d_in[8];
  const float* be2 = (const float*)d_in[9];
  const float* W3  = (const float*)d_in[10];
  const float* b3  = (const float*)d_in[11];
  const float* g3  = (const float*)d_in[12];
  const float* be3 = (const float*)d_in[13];
  const float* W4  = (const float*)d_in[14];
  const float* b4  = (const float*)d_in[15];

  const int Nn = in_sizes[0] / 24;
  const int E  = in_sizes[1] / 2;
  const int* src = ei;
  const int* dst = ei + E;

  // workspace layout (floats): dinv[N] | stats[256] | bufA[N*128] | bufB[N*128]
  float* ws    = (float*)d_ws;
  float* dinv  = ws;
  float* stats = ws + Nn;
  float* bufA  = stats + 256;
  float* bufB  = bufA + (size_t)Nn * 128;

  const float invN = 1.0f / (float)Nn;
  const int TB = 256;

  auto gemm = [&](const float* A, const float* Bm, float* C, int K, int F) {
    long waves = (long)((Nn + 15) / 16) * (F / 16);
    int blocks = (int)((waves * 32 + TB - 1) / TB);
    k_gemm_wmma_f32<<<blocks, TB, 0, stream>>>(A, Bm, dinv, C, Nn, K, F);
  };
  auto scatter = [&](const float* h, float* agg, int F) {
    int blocks = (int)(((long)E * 32 + TB - 1) / TB);
    k_scatter<<<blocks, TB, 0, stream>>>(h, src, dst, agg, E, F);
  };
  auto finish = [&](float* agg, const float* h, const float* bias, int F) {
    const int rows = 256;
    k_zero<<<1, 256, 0, stream>>>(stats, 256);
    k_finish_relu_stats<<<(Nn + rows - 1) / rows, F, 0, stream>>>(
        agg, h, dinv, bias, stats, Nn, F, rows);
  };
  auto bn = [&](float* v, const float* gamma, const float* beta, int F) {
    k_bn_apply<<<2048, TB, 0, stream>>>(v, stats, gamma, beta, Nn, F, invN);
  };
  auto zero = [&](float* p, size_t n) {
    int blocks = (int)((n + TB - 1) / TB);
    if (blocks > 4096) blocks = 4096;
    k_zero<<<blocks, TB, 0, stream>>>(p, n);
  };

  // ---- gcn_norm: dinv = rsqrt(indegree + 1) ----
  k_deg_init<<<(Nn + TB - 1) / TB, TB, 0, stream>>>(dinv, Nn);
  k_deg_count<<<(E + TB - 1) / TB, TB, 0, stream>>>(dst, dinv, E);
  k_rsqrt_inplace<<<(Nn + TB - 1) / TB, TB, 0, stream>>>(dinv, Nn);

  // ---- layer 1: x[N,24] -> 128 ----
  gemm(x, W1, bufA, 24, 128);
  zero(bufB, (size_t)Nn * 128);
  scatter(bufA, bufB, 128);
  finish(bufB, bufA, b1, 128);
  bn(bufB, g1, be1, 128);

  // ---- layer 2: 128 -> 128 ----
  gemm(bufB, W2, bufA, 128, 128);
  zero(bufB, (size_t)Nn * 128);
  scatter(bufA, bufB, 128);
  finish(bufB, bufA, b2, 128);
  bn(bufB, g2, be2, 128);

  // ---- layer 3: 128 -> 64 ----
  gemm(bufB, W3, bufA, 128, 64);
  zero(bufB, (size_t)Nn * 64);
  scatter(bufA, bufB, 64);
  finish(bufB, bufA, b3, 64);
  bn(bufB, g3, be3, 64);

  // ---- layer 4: 64 -> 2, then log_softmax ----
  k_gemm64x2<<<(Nn + TB - 1) / TB, TB, 0, stream>>>(bufB, W4, dinv, bufA, Nn);
  zero(bufB, (size_t)Nn * 2);
  k_scatter2<<<(E + TB - 1) / TB, TB, 0, stream>>>(bufA, src, dst, bufB, E);
  k_final_lsm<<<(Nn + TB - 1) / TB, TB, 0, stream>>>(bufB, bufA, dinv, b4,
                                                     (float*)d_out, Nn);
}